// CapsuleLayer_49641232007483
// MI455X (gfx1250) — compile-verified
//
#include <hip/hip_runtime.h>

typedef __attribute__((ext_vector_type(2))) float v2f;
typedef __attribute__((ext_vector_type(8))) float v8f;

namespace {
constexpr int B_TILE     = 16;    // batches per workgroup (= WMMA M)
constexpr int I_DIM      = 2048;
constexpr int D_DIM      = 8;     // WMMA K (2 x k4 steps)
constexpr int J_DIM      = 10;
constexpr int E_DIM      = 16;    // WMMA N
constexpr int NITER      = 3;
constexpr int WAVES      = 8;
constexpr int IT         = 2;     // i-columns in flight per wave
constexpr int NSPLIT     = 16;    // i-range splits -> 16 btiles x 16 splits = 256 WGs
constexpr int I_PER_WG   = I_DIM / NSPLIT;           // 128
constexpr int I_PER_WAVE = I_PER_WG / WAVES;         // 16 -> 8 i-steps per launch
constexpr int APAD       = 17;    // padded 16-float rows -> conflict-free banks
constexpr int UH_STRIDE  = 9;     // 8 accum floats + 1 pad per lane row
constexpr int OUT_ROW    = E_DIM + I_DIM;            // 2064
constexpr int BJ         = B_TILE * J_DIM;           // 160
constexpr int SEG        = BJ * E_DIM;               // 2560 floats per (btile|split) block

constexpr int UH_FLOATS  = WAVES * J_DIM * IT * 32 * UH_STRIDE; // 46080
constexpr int AS_FLOATS  = BJ * APAD;                           // 2720
constexpr int C_FLOATS   = WAVES * IT * BJ;                     // 2560
constexpr int LDS_FLOATS = UH_FLOATS + AS_FLOATS + WAVES * AS_FLOATS + C_FLOATS;
} // namespace

// One routing iteration over this WG's i-slice: u_hat via WMMA, logits from A,
// softmax over j, partial s accumulation -> deterministic write to Spart.
__global__ __launch_bounds__(WAVES * 32, 1)
void capsule_routing_kernel(const float* __restrict__ inp,    // [B, I, D]
                            const float* __restrict__ W,      // [J, I, E, D]
                            const float* __restrict__ Aglob,  // [btiles][b][j][e]
                            float* __restrict__ Spart,        // [btiles][split][b][j][e]
                            float* __restrict__ out,          // [B, J, E+I]
                            int last) {
  extern __shared__ float lds[];
  float* uh   = lds;                      // [w][j][it][lane][UH_STRIDE] native D-layout
  float* Aacc = uh + UH_FLOATS;           // [b][j][APAD]
  float* Spar = Aacc + AS_FLOATS;         // [w][b][j][APAD] per-wave partials
  float* Cbuf = Spar + WAVES * AS_FLOATS; // [w][it][b][j]

  const int tid   = threadIdx.x;
  const int wv    = tid >> 5;
  const int lane  = tid & 31;
  const int half  = lane >> 4;
  const int l16   = lane & 15;
  const int btile = blockIdx.x;
  const int split = blockIdx.y;

  // stage A_t into (padded) LDS
  for (int k = tid; k < SEG; k += WAVES * 32) {
    const int bj = k >> 4;
    const int e  = k & 15;
    Aacc[bj * APAD + e] = Aglob[(size_t)btile * SEG + k];
  }
  __syncthreads();

  const float* aBase =
      inp + ((size_t)(btile * B_TILE + l16) * I_DIM) * D_DIM + 2 * half;
  const int i0w = split * I_PER_WG + wv * I_PER_WAVE;

  // s accumulators in WMMA D-layout: s_reg[j][v] = s[b=v+8*half][j][e=l16]
  float s_reg[J_DIM][8];
#pragma unroll
  for (int j = 0; j < J_DIM; ++j)
#pragma unroll
    for (int v = 0; v < 8; ++v) s_reg[j][v] = 0.0f;

  for (int istep = 0; istep < I_PER_WAVE; istep += IT) {
    // ---------- Phase 1: u_hat tiles via V_WMMA_F32_16X16X4_F32 ----------
#pragma unroll
    for (int it = 0; it < IT; ++it) {
      const int i = i0w + istep + it;
      v2f a_lo = *(const v2f*)(aBase + (size_t)i * D_DIM);      // d = 0..3
      v2f a_hi = *(const v2f*)(aBase + (size_t)i * D_DIM + 4);  // d = 4..7
#pragma unroll
      for (int j = 0; j < J_DIM; ++j) {
        const float* wp =
            W + (((size_t)(j * I_DIM + i) * E_DIM) + l16) * D_DIM + 2 * half;
        v2f b_lo = *(const v2f*)wp;
        v2f b_hi = *(const v2f*)(wp + 4);
        v8f acc = {};
        acc = __builtin_amdgcn_wmma_f32_16x16x4_f32(
            false, a_lo, false, b_lo, (short)0, acc, false, false);
        acc = __builtin_amdgcn_wmma_f32_16x16x4_f32(
            false, a_hi, false, b_hi, (short)0, acc, false, false);
        float* dst =
            uh + (size_t)(((wv * J_DIM + j) * IT + it) * 32 + lane) * UH_STRIDE;
#pragma unroll
        for (int v = 0; v < 8; ++v) dst[v] = acc[v];
      }
    }
    asm volatile("s_wait_dscnt 0" ::: "memory");

    // ---------- Phase 2: logits -> softmax over j -> c ----------
    {
      const int b   = l16;   // each lane owns one (b, i) pair
      const int itp = half;
      const int i   = i0w + istep + itp;
      float logit[J_DIM];
#pragma unroll
      for (int j = 0; j < J_DIM; ++j) {
        const float* up =
            uh + (size_t)((((wv * J_DIM + j) * IT + itp) * 32) + 16 * (b >> 3)) *
                     UH_STRIDE + (b & 7);
        const float* ap = Aacc + (b * J_DIM + j) * APAD;
        float acc = 0.0f;
#pragma unroll
        for (int e = 0; e < E_DIM; ++e)
          acc = fmaf(ap[e], up[(size_t)e * UH_STRIDE], acc);
        logit[j] = acc;  // b_t[b,j,i] = A_t[b,j,:] . u_hat[b,j,i,:]
      }
      float m = logit[0];
#pragma unroll
      for (int j = 1; j < J_DIM; ++j) m = fmaxf(m, logit[j]);
      float cj[J_DIM];
      float ssum = 0.0f;
#pragma unroll
      for (int j = 0; j < J_DIM; ++j) {
        cj[j] = __expf(logit[j] - m);
        ssum += cj[j];
      }
      const float inv = 1.0f / ssum;
      float* cp = Cbuf + ((wv * IT + itp) * B_TILE + b) * J_DIM;
#pragma unroll
      for (int j = 0; j < J_DIM; ++j) {
        cj[j] *= inv;
        cp[j] = cj[j];
      }
      if (last) {  // stream final coupling coeffs: out[b, j, 16 + i]
        float* op =
            out + (size_t)(btile * B_TILE + b) * J_DIM * OUT_ROW + E_DIM + i;
#pragma unroll
        for (int j = 0; j < J_DIM; ++j) op[(size_t)j * OUT_ROW] = cj[j];
      }
    }
    asm volatile("s_wait_dscnt 0" ::: "memory");

    // ---------- Phase 3: s[j][b,e] += c[b,j,i] * u_hat (D-layout) ----------
#pragma unroll
    for (int it = 0; it < IT; ++it) {
#pragma unroll
      for (int j = 0; j < J_DIM; ++j) {
        const float* up =
            uh + (size_t)((((wv * J_DIM + j) * IT + it) * 32) + lane) * UH_STRIDE;
        const float* cp = Cbuf + (wv * IT + it) * BJ + j;
#pragma unroll
        for (int v = 0; v < 8; ++v) {
          const float c = cp[(v + 8 * half) * J_DIM];
          s_reg[j][v] = fmaf(c, up[v], s_reg[j][v]);
        }
      }
    }
  }  // istep

  // deterministic cross-wave reduction: dump per-wave partials
  {
    float* sp = Spar + wv * AS_FLOATS;
#pragma unroll
    for (int j = 0; j < J_DIM; ++j)
#pragma unroll
      for (int v = 0; v < 8; ++v) {
        const int b = v + 8 * half;
        sp[(b * J_DIM + j) * APAD + l16] = s_reg[j][v];
      }
  }
  __syncthreads();

  // reduce 8 wave partials in fixed order, write this split's partial s
  if (tid < BJ) {
    const int row = tid * APAD;
    float* dst = Spart + ((size_t)btile * NSPLIT + split) * SEG + tid * E_DIM;
#pragma unroll
    for (int e = 0; e < E_DIM; ++e) {
      float x = 0.0f;
#pragma unroll
      for (int w = 0; w < WAVES; ++w) x += Spar[w * AS_FLOATS + row + e];
      dst[e] = x;
    }
  }
}

// Reduce split partials (fixed order), squash, update A (or emit outputs).
__global__ __launch_bounds__(BJ, 1)
void capsule_squash_kernel(const float* __restrict__ Spart,
                           float* __restrict__ Aglob,
                           float* __restrict__ out,
                           int last) {
  const int btile = blockIdx.x;
  const int tid   = threadIdx.x;  // (b*J + j)
  const float* sp = Spart + (size_t)btile * NSPLIT * SEG + tid * E_DIM;

  float v[E_DIM];
  float ss = 0.0f;
#pragma unroll
  for (int e = 0; e < E_DIM; ++e) {
    float x = 0.0f;
#pragma unroll
    for (int s = 0; s < NSPLIT; ++s) x += sp[(size_t)s * SEG + e];
    v[e] = x;
    ss = fmaf(x, x, ss);
  }
  const float scale = ss / (1.0f + ss) * __frsqrt_rn(ss + 1e-7f);  // squash
  if (last) {
    float* op = out + ((size_t)btile * BJ + tid) * OUT_ROW;
#pragma unroll
    for (int e = 0; e < E_DIM; ++e) op[e] = scale * v[e];
  } else {
    float* ap = Aglob + (size_t)btile * SEG + tid * E_DIM;
#pragma unroll
    for (int e = 0; e < E_DIM; ++e) ap[e] += scale * v[e];  // A_{t+1} = A_t + out_t
  }
}

extern "C" void kernel_launch(void* const* d_in, const int* in_sizes, int n_in,
                              void* d_out, int out_size, void* d_ws, size_t ws_size,
                              hipStream_t stream) {
  const float* inputs = (const float*)d_in[0];  // [B, I, D] f32
  const float* W      = (const float*)d_in[1];  // [J, I, E, D] f32
  float* out          = (float*)d_out;          // [B, J, E+I] f32

  const int B      = in_sizes[0] / (I_DIM * D_DIM);  // 256
  const int btiles = B / B_TILE;                     // 16

  float* Aglob = (float*)d_ws;                       // btiles * SEG floats (160 KB)
  float* Spart = Aglob + (size_t)btiles * SEG;       // btiles * NSPLIT * SEG (2.6 MB)

  const size_t lds_bytes = (size_t)LDS_FLOATS * sizeof(float);  // ~292 KB < 320 KB/WGP
  (void)hipFuncSetAttribute((const void*)capsule_routing_kernel,
                            hipFuncAttributeMaxDynamicSharedMemorySize,
                            (int)lds_bytes);

  // A_0 = 0 (re-zeroed every call for determinism)
  (void)hipMemsetAsync(Aglob, 0, (size_t)btiles * SEG * sizeof(float), stream);

  for (int t = 0; t < NITER; ++t) {
    const int last = (t == NITER - 1);
    capsule_routing_kernel<<<dim3(btiles, NSPLIT), dim3(WAVES * 32), lds_bytes,
                             stream>>>(inputs, W, Aglob, Spart, out, last);
    capsule_squash_kernel<<<dim3(btiles), dim3(BJ), 0, stream>>>(Spart, Aglob,
                                                                 out, last);
  }
}